// GAT_57062935495232
// MI455X (gfx1250) — compile-verified
//
#include <hip/hip_runtime.h>

// -----------------------------------------------------------------------------
// 3-layer GAT forward for MI455X (gfx1250, wave32).
//   GEMMs     : V_WMMA_F32_16X16X4_F32; A-tile staged into LDS by the Tensor
//               Data Mover (tensor_load_to_lds + s_wait_tensorcnt), B from L2.
//   Softmax   : segment max via int-trick atomic max, segment sum via f32 atomics
//   Aggregate : one wave per (edge, head), coalesced b64 loads,
//               global_atomic_add_f32 scatter.
// Working set (~60 MB) is L2-resident (192 MB).
// -----------------------------------------------------------------------------

namespace {
constexpr int NN     = 20000;           // nodes
constexpr int NE     = 640000;          // edges (before self loops)
constexpr int ETOT   = NE + NN;         // edges incl. self loops
constexpr int IN_F   = 128;
constexpr int HIDC   = 64;
constexpr int NHEADS = 4;
constexpr int OUT_F  = 64;
constexpr float SLOPE = 0.2f;
constexpr float EPSV  = 1e-16f;
}

typedef __attribute__((ext_vector_type(2))) float        v2f;
typedef __attribute__((ext_vector_type(8))) float        v8f;
typedef __attribute__((ext_vector_type(4))) unsigned int v4u;
typedef __attribute__((ext_vector_type(8))) int          v8i;
typedef __attribute__((ext_vector_type(4))) int          v4i;

// ---------------------------------------------------------------- fill -------
__global__ void fill_f32(float* __restrict__ p, float v, int count) {
  int i = blockIdx.x * blockDim.x + threadIdx.x;
  if (i < count) p[i] = v;
}

// ------------------------------------------------- TDM 2-D f32 tile -> LDS ---
// D# per CDNA5 ISA §8.3/§8.4. 2-D tile: groups 2/3 zero (tile_dim2 = 0).
//   group0: [1:0]count=1 | [63:32]lds_addr | [120:64]global_addr | [127:126]=2
//   group1: [17:16]data_size=2(4B) | tensor_dim0/1 | tile_dim0/1 | dim0 stride
// clang-23 / therock-10.0 builtin: 6 args (v4u, v8i, v4i, v4i, v8i, i32 cpol).
__device__ __forceinline__ void tdm_load_tile_f32(unsigned lds_byte_off,
                                                  const float* gptr,
                                                  unsigned tile_w,     // elems
                                                  unsigned tile_h,     // rows
                                                  unsigned row_stride) // elems
{
  unsigned long long ga = (unsigned long long)(size_t)gptr;
  v4u g0;
  g0[0] = 1u;                                   // count=1 (valid user D#)
  g0[1] = lds_byte_off;                         // lds_addr (bytes)
  g0[2] = (unsigned)ga;                         // global_addr[31:0]
  g0[3] = (unsigned)((ga >> 32) & 0x01FFFFFFu)  // global_addr[56:32]
        | (2u << 30);                           // type=2 ("image")
  v8i g1;
  g1[0] = (int)(2u << 16);                      // wg_mask=0, data_size=4B
  g1[1] = (int)((tile_w & 0xFFFFu) << 16);      // tensor_dim0[15:0]
  g1[2] = (int)((tile_w >> 16) | ((tile_h & 0xFFFFu) << 16)); // td0 hi | td1 lo
  g1[3] = (int)((tile_h >> 16) | ((tile_w & 0xFFFFu) << 16)); // td1 hi | tile_dim0
  g1[4] = (int)(tile_h & 0xFFFFu);              // tile_dim1 | tile_dim2=0
  g1[5] = (int)row_stride;                      // tensor_dim0_stride[31:0]
  g1[6] = 0;                                    // stride hi | dim1_stride lo
  g1[7] = 0;                                    // dim1_stride hi
  v4i gz4 = {0, 0, 0, 0};
  v8i gz8 = {0, 0, 0, 0, 0, 0, 0, 0};
  __builtin_amdgcn_tensor_load_to_lds(g0, g1, gz4, gz4, gz8, 0);
}

// ------------------------------------------------------- WMMA fp32 GEMM ------
// C[M,kout] = X[M,kin] * W[kin,kout], M % 16 == 0, kout % (16*waves) == 0.
// Wave 0 DMAs the shared 16 x kin A-tile into LDS via the TDM; every wave then
// owns one 16x16 C tile, K advancing 4 per V_WMMA_F32_16X16X4_F32.
//
// A (16x4 f32): lanes 0-15 -> M=0..15, {v0,v1}=K{0,1}; lanes 16-31 K{2,3}.
// B (4x16 f32): lanes 0-15 -> N=0..15, {v0,v1}=K{0,1}; lanes 16-31 K{2,3}.
// C/D (16x16 f32, 8 VGPRs): vgpr r, lanes 0-15 -> M=r; lanes 16-31 -> M=r+8.
__global__ void gemm_wmma_f32(const float* __restrict__ X,
                              const float* __restrict__ W,
                              float* __restrict__ C,
                              int kin, int kout) {
  __shared__ float tileA[16 * 256];            // 16 rows x kin<=256 cols, 16 KB
  const int wave = threadIdx.x >> 5;
  const int lane = threadIdx.x & 31;
  const int half = lane >> 4;                  // which K-pair this half holds
  const int l16  = lane & 15;
  const int row0 = blockIdx.x * 16;
  const int col  = (blockIdx.y * (blockDim.x >> 5) + wave) * 16 + l16;

  if (wave == 0) {
    // flat->LDS aperture: addr[31:0] is the LDS byte offset
    tdm_load_tile_f32((unsigned)(size_t)(void*)tileA,
                      X + (size_t)row0 * kin,
                      (unsigned)kin, 16u, (unsigned)kin);
    __builtin_amdgcn_s_wait_tensorcnt(0);
  }
  __syncthreads();

  const float* __restrict__ ar = tileA + l16 * kin;
  v8f acc = {};
#pragma unroll 4
  for (int k = 0; k < kin; k += 4) {
    const int ka = k + half * 2;
    v2f a, b;
    a.x = ar[ka];                               // ds_load_b64 from staged tile
    a.y = ar[ka + 1];
    b.x = W[(size_t)ka * kout + col];
    b.y = W[(size_t)(ka + 1) * kout + col];
    // 8-arg form: (neg_a, A, neg_b, B, c_mod, C, reuse_a, reuse_b)
    acc = __builtin_amdgcn_wmma_f32_16x16x4_f32(false, a, false, b,
                                                (short)0, acc, false, false);
  }

  float* __restrict__ cr = C + (size_t)(row0 + half * 8) * kout + col;
#pragma unroll
  for (int r = 0; r < 8; ++r) cr[(size_t)r * kout] = acc[r];
}

// --------------------------------------------------- per-node attn logits ----
__global__ void node_logits(const float* __restrict__ h,
                            const float* __restrict__ a_s,
                            const float* __restrict__ a_d,
                            float* __restrict__ als,
                            float* __restrict__ ald,
                            int n, int heads, int ch) {
  int i = blockIdx.x * blockDim.x + threadIdx.x;
  if (i >= n * heads) return;
  int node = i / heads;
  int hd = i - node * heads;
  const float* __restrict__ hp = h + ((size_t)node * heads + hd) * ch;
  const float* __restrict__ sp = a_s + (size_t)hd * ch;
  const float* __restrict__ dp = a_d + (size_t)hd * ch;
  float s = 0.f, d = 0.f;
#pragma unroll 4
  for (int c = 0; c < ch; ++c) {
    s = fmaf(hp[c], sp[c], s);
    d = fmaf(hp[c], dp[c], d);
  }
  als[i] = s;
  ald[i] = d;
}

// ------------------------------------------------------------- edge utils ----
__device__ __forceinline__ void atomicMaxF32(float* addr, float val) {
  // Monotone int/uint encoding of IEEE floats (init value is -inf).
  if (val >= 0.0f) atomicMax((int*)addr, __float_as_int(val));
  else             atomicMin((unsigned int*)addr, __float_as_uint(val));
}

__device__ __forceinline__ void edge_ends(const int* __restrict__ ei, int e,
                                          int& s, int& d) {
  if (e < NE) { s = ei[e]; d = ei[NE + e]; }
  else        { s = e - NE; d = e - NE; }      // self loop
}

__device__ __forceinline__ float edge_logit(const float* __restrict__ als,
                                            const float* __restrict__ ald,
                                            int s, int d, int heads, int hd) {
  float x = als[s * heads + hd] + ald[d * heads + hd];
  return x > 0.0f ? x : SLOPE * x;             // leaky_relu(0.2)
}

// ------------------------------------------------------- segment max ---------
__global__ void edge_max(const int* __restrict__ ei,
                         const float* __restrict__ als,
                         const float* __restrict__ ald,
                         float* __restrict__ m, int heads) {
  int i = blockIdx.x * blockDim.x + threadIdx.x;
  if (i >= ETOT * heads) return;
  int e = i / heads, hd = i - e * heads;
  int s, d; edge_ends(ei, e, s, d);
  atomicMaxF32(&m[d * heads + hd], edge_logit(als, ald, s, d, heads, hd));
}

// ------------------------------------------------------- segment exp-sum -----
__global__ void edge_expsum(const int* __restrict__ ei,
                            const float* __restrict__ als,
                            const float* __restrict__ ald,
                            const float* __restrict__ m,
                            float* __restrict__ ssum, int heads) {
  int i = blockIdx.x * blockDim.x + threadIdx.x;
  if (i >= ETOT * heads) return;
  int e = i / heads, hd = i - e * heads;
  int s, d; edge_ends(ei, e, s, d);
  float x = edge_logit(als, ald, s, d, heads, hd);
  atomicAdd(&ssum[d * heads + hd], __expf(x - m[d * heads + hd]));
}

// ------------------------------------------------- alpha-weighted scatter ----
// grid.x = ETOT, block = heads*32; wave hd scatters alpha * h[src,hd,:] into
// out[dst,hd,:]: one coalesced float2 load + two global_atomic_add_f32 / lane.
__global__ void edge_aggregate(const int* __restrict__ ei,
                               const float* __restrict__ als,
                               const float* __restrict__ ald,
                               const float* __restrict__ m,
                               const float* __restrict__ ssum,
                               const float* __restrict__ hfeat,
                               float* __restrict__ out,
                               int heads, int ch) {
  int e = blockIdx.x;
  int hd = threadIdx.x >> 5;
  int lane = threadIdx.x & 31;
  int s, d; edge_ends(ei, e, s, d);
  const float* __restrict__ hp = hfeat + ((size_t)s * heads + hd) * ch;
  __builtin_prefetch(hp, 0, 0);                // global_prefetch_b8 on src row
  float x = edge_logit(als, ald, s, d, heads, hd);
  float alpha = __expf(x - m[d * heads + hd]) / (ssum[d * heads + hd] + EPSV);
  float* __restrict__ op = out + ((size_t)d * heads + hd) * ch;
  for (int c = lane * 2; c < ch; c += 64) {
    float2 hv = *reinterpret_cast<const float2*>(hp + c);  // global_load_b64
    atomicAdd(&op[c + 0], alpha * hv.x);
    atomicAdd(&op[c + 1], alpha * hv.y);
  }
}

// -------------------------------------------------------- bias (+ReLU) ------
__global__ void bias_act(float* __restrict__ x, const float* __restrict__ b,
                         int total, int f, int do_relu) {
  int i = blockIdx.x * blockDim.x + threadIdx.x;
  if (i >= total) return;
  float v = x[i] + b[i % f];
  x[i] = do_relu ? fmaxf(v, 0.0f) : v;
}

// -----------------------------------------------------------------------------
extern "C" void kernel_launch(void* const* d_in, const int* in_sizes, int n_in,
                              void* d_out, int out_size, void* d_ws, size_t ws_size,
                              hipStream_t stream) {
  (void)in_sizes; (void)n_in; (void)out_size; (void)ws_size;

  const float* x   = (const float*)d_in[0];
  const int*   ei  = (const int*)d_in[1];
  const float* W0  = (const float*)d_in[2];
  const float* as0 = (const float*)d_in[3];
  const float* ad0 = (const float*)d_in[4];
  const float* b0  = (const float*)d_in[5];
  const float* W1  = (const float*)d_in[6];
  const float* as1 = (const float*)d_in[7];
  const float* ad1 = (const float*)d_in[8];
  const float* b1  = (const float*)d_in[9];
  const float* W2  = (const float*)d_in[10];
  const float* as2 = (const float*)d_in[11];
  const float* ad2 = (const float*)d_in[12];
  const float* b2  = (const float*)d_in[13];
  float* out = (float*)d_out;

  // ---- workspace layout (floats): 2*N*256 + 4*N*H  ~= 42.2 MB ----
  float* hbuf   = (float*)d_ws;                     // [NN, 256]  h = X @ W
  float* aggbuf = hbuf   + (size_t)NN * 256;        // [NN, 256]  aggregated
  float* als    = aggbuf + (size_t)NN * 256;        // [NN, H]
  float* ald    = als    + (size_t)NN * NHEADS;     // [NN, H]
  float* mmax   = ald    + (size_t)NN * NHEADS;     // [NN, H]
  float* ssum   = mmax   + (size_t)NN * NHEADS;     // [NN, H]

  const float NEG_INF = -__builtin_inff();
  const int FW = NHEADS * HIDC;                     // 256
  auto cdiv = [](int a, int b) { return (a + b - 1) / b; };

  // =========================== Layer 0: 128 -> 4x64 (concat, ReLU) ==========
  {
    dim3 grid(NN / 16, FW / (16 * 8));              // block = 256 -> 8 waves
    gemm_wmma_f32<<<grid, 256, 0, stream>>>(x, W0, hbuf, IN_F, FW);
    const int nh = NN * NHEADS;
    node_logits<<<cdiv(nh, 256), 256, 0, stream>>>(hbuf, as0, ad0, als, ald,
                                                   NN, NHEADS, HIDC);
    fill_f32<<<cdiv(nh, 256), 256, 0, stream>>>(mmax, NEG_INF, nh);
    fill_f32<<<cdiv(nh, 256), 256, 0, stream>>>(ssum, 0.0f, nh);
    fill_f32<<<cdiv(NN * FW, 256), 256, 0, stream>>>(aggbuf, 0.0f, NN * FW);
    edge_max<<<cdiv(ETOT * NHEADS, 256), 256, 0, stream>>>(ei, als, ald, mmax, NHEADS);
    edge_expsum<<<cdiv(ETOT * NHEADS, 256), 256, 0, stream>>>(ei, als, ald, mmax, ssum, NHEADS);
    edge_aggregate<<<ETOT, NHEADS * 32, 0, stream>>>(ei, als, ald, mmax, ssum,
                                                     hbuf, aggbuf, NHEADS, HIDC);
    bias_act<<<cdiv(NN * FW, 256), 256, 0, stream>>>(aggbuf, b0, NN * FW, FW, 1);
  }

  // =========================== Layer 1: 256 -> 4x64 (concat, ReLU) ==========
  {
    dim3 grid(NN / 16, FW / (16 * 8));
    gemm_wmma_f32<<<grid, 256, 0, stream>>>(aggbuf, W1, hbuf, FW, FW);
    const int nh = NN * NHEADS;
    node_logits<<<cdiv(nh, 256), 256, 0, stream>>>(hbuf, as1, ad1, als, ald,
                                                   NN, NHEADS, HIDC);
    fill_f32<<<cdiv(nh, 256), 256, 0, stream>>>(mmax, NEG_INF, nh);
    fill_f32<<<cdiv(nh, 256), 256, 0, stream>>>(ssum, 0.0f, nh);
    // safe: the GEMM above (same stream) already consumed aggbuf
    fill_f32<<<cdiv(NN * FW, 256), 256, 0, stream>>>(aggbuf, 0.0f, NN * FW);
    edge_max<<<cdiv(ETOT * NHEADS, 256), 256, 0, stream>>>(ei, als, ald, mmax, NHEADS);
    edge_expsum<<<cdiv(ETOT * NHEADS, 256), 256, 0, stream>>>(ei, als, ald, mmax, ssum, NHEADS);
    edge_aggregate<<<ETOT, NHEADS * 32, 0, stream>>>(ei, als, ald, mmax, ssum,
                                                     hbuf, aggbuf, NHEADS, HIDC);
    bias_act<<<cdiv(NN * FW, 256), 256, 0, stream>>>(aggbuf, b1, NN * FW, FW, 1);
  }

  // =========================== Layer 2: 256 -> 1x64 (mean==identity) ========
  {
    dim3 grid(NN / 16, OUT_F / (16 * 4));           // block = 128 -> 4 waves
    gemm_wmma_f32<<<grid, 128, 0, stream>>>(aggbuf, W2, hbuf, FW, OUT_F);
    node_logits<<<cdiv(NN, 256), 256, 0, stream>>>(hbuf, as2, ad2, als, ald,
                                                   NN, 1, OUT_F);
    fill_f32<<<cdiv(NN, 256), 256, 0, stream>>>(mmax, NEG_INF, NN);
    fill_f32<<<cdiv(NN, 256), 256, 0, stream>>>(ssum, 0.0f, NN);
    fill_f32<<<cdiv(NN * OUT_F, 256), 256, 0, stream>>>(out, 0.0f, NN * OUT_F);
    edge_max<<<cdiv(ETOT, 256), 256, 0, stream>>>(ei, als, ald, mmax, 1);
    edge_expsum<<<cdiv(ETOT, 256), 256, 0, stream>>>(ei, als, ald, mmax, ssum, 1);
    edge_aggregate<<<ETOT, 32, 0, stream>>>(ei, als, ald, mmax, ssum,
                                            hbuf, out, 1, OUT_F);
    bias_act<<<cdiv(NN * OUT_F, 256), 256, 0, stream>>>(out, b2, NN * OUT_F, OUT_F, 0);
  }
}